// QubitContextLookahead_82008105550026
// MI455X (gfx1250) — compile-verified
//
#include <hip/hip_runtime.h>

typedef float v2f __attribute__((ext_vector_type(2)));
typedef float v8f __attribute__((ext_vector_type(8)));

#define LW   0.5f
#define BDIM 16
#define TDIM 256
#define QDIM 64
#define DDIM 128
#define AS   68     // LDS row stride (floats) for 64x64 adjacency  (4l mod 64 bank spread)
#define XS   132    // LDS row stride (floats) for 64x128 activations (528B rows, 16B aligned)
#define YS   132    // LDS row stride (float2) for packed row-pair Y buffer

// ---------------------------------------------------------------------------
// Kernel 1: backward lookahead scan over T; store weights with diagonal = 1.
// One block per (b, q) row; 64 threads = columns; coalesced 256B per step.
// Carry keeps the un-patched scan value (reference sets diag only inside GCN).
// ---------------------------------------------------------------------------
__global__ void lookahead_scan(const float* __restrict__ adj,
                               float* __restrict__ wbuf) {
  const int b = blockIdx.x >> 6;
  const int q = blockIdx.x & 63;
  const int c = threadIdx.x;                 // 0..63
  const bool isDiag = (c == q);
  float carry = 0.0f;
  const size_t base = ((size_t)b * TDIM) * (QDIM * QDIM) + (size_t)q * QDIM + c;
  for (int t = TDIM - 1; t >= 0; --t) {
    const size_t idx = base + (size_t)t * (QDIM * QDIM);
    const float v = adj[idx];
    carry = LW * (carry + v);
    wbuf[idx] = isDiag ? 1.0f : carry;
  }
}

// ---------------------------------------------------------------------------
// Kernel 2: pack a 128x128 weight matrix into WMMA B-fragment order:
//   Wp[k4][col][p] = { W[4k4+2p][col], W[4k4+2p+1][col] }   (float2)
// so each lane's B fragment is one aligned 8B load, 256B contiguous per wave.
// ---------------------------------------------------------------------------
__global__ void pack_weight(const float* __restrict__ W,
                            float* __restrict__ Wp) {
  const int idx = blockIdx.x * blockDim.x + threadIdx.x;  // 0..8191
  const int k4  = idx >> 8;
  const int col = (idx & 255) >> 1;
  const int p   = idx & 1;
  const int kr  = 4 * k4 + 2 * p;
  Wp[2 * idx]     = W[kr * DDIM + col];
  Wp[2 * idx + 1] = W[(kr + 1) * DDIM + col];
}

// ---------------------------------------------------------------------------
// Kernel 3: fused 2-layer dense GCN per graph.
// 256 threads = 8 wave32; wave (r,g) owns rows [16r,16r+16) x cols [64g,64g+64).
// All matmuls via V_WMMA_F32_16X16X4_F32 (f32-exact, matches reference dtype).
// ---------------------------------------------------------------------------
__global__ __launch_bounds__(256)
void gcn_fused(const float* __restrict__ xin,
               const float* __restrict__ wbuf,
               const float* __restrict__ Wp1, const float* __restrict__ b1,
               const float* __restrict__ Wp2, const float* __restrict__ b2,
               float* __restrict__ out) {
  __shared__ float sA[QDIM * AS];          // normalized adjacency 64x64
  __shared__ float sX[QDIM * XS];          // activations 64x128 (row major)
  __shared__ float sY[(QDIM / 2) * YS * 2];// X@W result, packed row-pairs (float2)
  __shared__ float sDis[QDIM];

  const int tid  = threadIdx.x;
  const int lane = tid & 31;
  const int wv   = tid >> 5;               // 0..7
  const int r    = wv & 3;                 // row strip
  const int g    = wv >> 2;                // column half
  const int lm   = lane & 15;
  const int p    = lane >> 4;              // lane half -> K sub-pair select
  const size_t n = blockIdx.x;             // graph id in [0, 4096)

  // ---- load adjacency (diag already 1) ----
  {
    const float* wsrc = wbuf + n * (size_t)(QDIM * QDIM);
#pragma unroll
    for (int i = 0; i < 16; ++i) {
      const int f  = i * 256 + tid;
      const int rr = f >> 6, cc = f & 63;
      sA[rr * AS + cc] = wsrc[f];
    }
  }
  // ---- load X (float4, coalesced; 528B LDS rows keep 16B alignment) ----
  {
    const float4* xsrc = (const float4*)(xin + n * (size_t)(QDIM * DDIM));
#pragma unroll
    for (int i = 0; i < 8; ++i) {
      const int f4 = i * 256 + tid;
      const int rr = f4 >> 5, c4 = f4 & 31;
      *(float4*)&sX[rr * XS + c4 * 4] = xsrc[f4];
    }
  }
  __syncthreads();

  // ---- symmetric degree normalization: A = D^-1/2 (W+I) D^-1/2 ----
  if (tid < QDIM) {
    float s = 0.0f;
#pragma unroll 4
    for (int cc = 0; cc < QDIM; ++cc) s += sA[tid * AS + cc];
    sDis[tid] = rsqrtf(fmaxf(s, 1.0f));
  }
  __syncthreads();
#pragma unroll
  for (int i = 0; i < 16; ++i) {
    const int f  = i * 256 + tid;
    const int rr = f >> 6, cc = f & 63;
    sA[rr * AS + cc] *= sDis[rr] * sDis[cc];
  }
  __syncthreads();

  // GEMM Y = sIn(64x128) @ W(128x128), W pre-packed; result -> sY row-pairs.
  auto gemm_xw = [&](const float* __restrict__ sIn, const float* __restrict__ Wp) {
#pragma unroll
    for (int c = 0; c < 4; ++c) {
      const int colbase = g * 64 + c * 16;
      v8f acc = {0.f, 0.f, 0.f, 0.f, 0.f, 0.f, 0.f, 0.f};
      const float* arow = sIn + (r * 16 + lm) * XS + 2 * p;
      const float* bcol = Wp + 2 * ((colbase + lm) * 2 + p);
#pragma unroll
      for (int k4 = 0; k4 < 32; ++k4) {
        v2f a = *(const v2f*)(arow + 4 * k4);     // A[lm][k+2p], A[lm][k+2p+1]
        v2f b = *(const v2f*)(bcol + 512 * k4);   // W[k+2p][col], W[k+2p+1][col]
        acc = __builtin_amdgcn_wmma_f32_16x16x4_f32(
            false, a, false, b, (short)0, acc, false, false);
      }
      // store as row-pairs: rows (2m,2m+1) -> float2, ready as next B fragments
#pragma unroll
      for (int j2 = 0; j2 < 4; ++j2) {
        const int rp = r * 8 + p * 4 + j2;
        v2f d;
        d.x = acc[2 * j2];
        d.y = acc[2 * j2 + 1];
        *(v2f*)&sY[2 * (rp * YS + colbase + lm)] = d;
      }
    }
  };

  // GEMM Z = sA(64x64) @ Y(64x128) + bias; result -> sOut row-major.
  auto gemm_ay = [&](const float* __restrict__ bias, float* __restrict__ sOut) {
#pragma unroll
    for (int c = 0; c < 4; ++c) {
      const int colbase = g * 64 + c * 16;
      const float bb = bias[colbase + lm];
      v8f acc = {bb, bb, bb, bb, bb, bb, bb, bb};
      const float* arow = sA + (r * 16 + lm) * AS + 2 * p;
      const float* bcol = sY + 2 * ((size_t)p * YS + colbase + lm);
#pragma unroll
      for (int k4 = 0; k4 < 16; ++k4) {
        v2f a = *(const v2f*)(arow + 4 * k4);     // A[lm][k+2p..k+2p+1]
        v2f b = *(const v2f*)(bcol + 528 * k4);   // Y[k+2p][col], Y[k+2p+1][col]
        acc = __builtin_amdgcn_wmma_f32_16x16x4_f32(
            false, a, false, b, (short)0, acc, false, false);
      }
#pragma unroll
      for (int j = 0; j < 8; ++j)
        sOut[(r * 16 + p * 8 + j) * XS + colbase + lm] = acc[j];
    }
  };

  gemm_xw(sX, Wp1);  __syncthreads();   // Y1 = X @ W1
  gemm_ay(b1, sX);   __syncthreads();   // Z1 = A @ Y1 + b1   (overwrites X)
  gemm_xw(sX, Wp2);  __syncthreads();   // Y2 = Z1 @ W2
  gemm_ay(b2, sX);   __syncthreads();   // Z2 = A @ Y2 + b2   (staged in LDS)

  // ---- coalesced float4 writeback ----
  {
    float4* dst = (float4*)(out + n * (size_t)(QDIM * DDIM));
#pragma unroll
    for (int i = 0; i < 8; ++i) {
      const int f4 = i * 256 + tid;
      const int rr = f4 >> 5, c4 = f4 & 31;
      dst[f4] = *(const float4*)&sX[rr * XS + c4 * 4];
    }
  }
}

extern "C" void kernel_launch(void* const* d_in, const int* in_sizes, int n_in,
                              void* d_out, int out_size, void* d_ws, size_t ws_size,
                              hipStream_t stream) {
  const float* x   = (const float*)d_in[0];  // [16,256,64,128]
  const float* adj = (const float*)d_in[1];  // [16,256,64,64]
  const float* W1  = (const float*)d_in[2];  // [128,128]
  const float* b1  = (const float*)d_in[3];  // [128]
  const float* W2  = (const float*)d_in[4];  // [128,128]
  const float* b2  = (const float*)d_in[5];  // [128]
  float* out = (float*)d_out;

  // workspace layout: packed W1 (64KB) | packed W2 (64KB) | lookahead weights (67MB)
  float* Wp1  = (float*)d_ws;
  float* Wp2  = Wp1 + 16384;
  float* wbuf = Wp2 + 16384;

  lookahead_scan<<<BDIM * QDIM, 64, 0, stream>>>(adj, wbuf);
  pack_weight<<<32, 256, 0, stream>>>(W1, Wp1);
  pack_weight<<<32, 256, 0, stream>>>(W2, Wp2);
  gcn_fused<<<BDIM * TDIM, 256, 0, stream>>>(x, wbuf, Wp1, b1, Wp2, b2, out);
}